// MemoryGraph_55757265437318
// MI455X (gfx1250) — compile-verified
//
#include <hip/hip_runtime.h>
#include <math.h>

// Problem constants (must match the reference)
#define BSb   4
#define TSEG  16
#define CCc   64
#define DDd   64
#define NNn   4096
#define KKk   32
#define HHh   64
#define EPSF  1e-8f

typedef __attribute__((ext_vector_type(2))) float v2f;
typedef __attribute__((ext_vector_type(8))) float v8f;

__device__ __forceinline__ float sigf(float x) { return 1.0f / (1.0f + expf(-x)); }

// ---------------------------------------------------------------------------
// Kernel 1: per-node modulator. One wave32 per node n.
//   - builds mod_input (4 x 320) in LDS
//   - fc1 GEMM via V_WMMA_F32_16X16X4_F32  (M=4 valid rows of 16, K=320, N=64)
//   - fc2 (64x3), gates, eff_prim / eff_key / eff_decay, routing
// ---------------------------------------------------------------------------
__global__ void __launch_bounds__(256) k_mod(
    const float* __restrict__ h_prev, const float* __restrict__ prev_msg,
    const float* __restrict__ trace_prim, const float* __restrict__ trace_key,
    const float* __restrict__ primitives, const float* __restrict__ key_p,
    const float* __restrict__ decay_logit,
    const float* __restrict__ fc1_w, const float* __restrict__ fc1_b,
    const float* __restrict__ fc2_w, const float* __restrict__ fc2_b,
    const float* __restrict__ mod_lr_logit, const int* __restrict__ conn,
    float* __restrict__ eff_prim, float* __restrict__ eff_decay,
    float* __restrict__ routing)
{
    __shared__ float s_mi[8][1280];   // mod_input [b][320]
    __shared__ float s_xs[8][256];    // x = tanh(fc1) [b][64]
    __shared__ float s_ek[8][256];    // eff_key [b][64]
    __shared__ float s_o3[8][12];     // out3 [b][3]
    __shared__ float s_nrm[8][8];     // ||trace_prim||, ||trace_key|| per b
    __shared__ float s_sc[8][8];      // scale factors per b

    const int w    = threadIdx.x >> 5;
    const int lane = threadIdx.x & 31;
    const int n    = blockIdx.x * 8 + w;

    float* mi  = s_mi[w];
    float* xs  = s_xs[w];
    float* ek  = s_ek[w];
    float* o3  = s_o3[w];
    float* nrm = s_nrm[w];
    float* sc  = s_sc[w];

    // ---- stage mod_input = [h_prev | trace_prim | trace_key | primitives | key_p]
    const size_t nD = (size_t)n * DDd;
    for (int i = lane; i < BSb * 320; i += 32) {
        int b = i / 320;
        int dIn = i - b * 320;
        int seg = dIn >> 6;
        int d = dIn & 63;
        size_t gi = ((size_t)b * NNn + n) * DDd + d;
        float v;
        if      (seg == 0) v = h_prev[gi];
        else if (seg == 1) v = trace_prim[gi];
        else if (seg == 2) v = trace_key[gi];
        else if (seg == 3) v = primitives[nD + d];
        else               v = key_p[nD + d];
        mi[i] = v;
    }
    __syncthreads();

    // ---- trace norms (lanes 0..7: b = lane&3, which = lane>>2)
    if (lane < 8) {
        int b = lane & 3;
        int off = 64 + (lane >> 2) * 64;
        float s = 0.f;
        #pragma unroll 8
        for (int d = 0; d < 64; ++d) { float v = mi[b * 320 + off + d]; s += v * v; }
        nrm[lane] = sqrtf(s);
    }
    __syncthreads();

    // ---- fc1 GEMM via WMMA f32 16x16x4. Four 16-column tiles, K loop of 80.
    // A layout (16x4 f32): VGPR0 = K+0 (lanes0-15) / K+2 (lanes16-31); VGPR1 = K+1 / K+3
    // B layout (4x16 f32): VGPR0 = rows K+0 / K+2 striped over N in lanes; VGPR1 = K+1 / K+3
    v8f acc0 = {0.f,0.f,0.f,0.f,0.f,0.f,0.f,0.f};
    v8f acc1 = acc0, acc2 = acc0, acc3 = acc0;

    const int row = lane & 15;         // M index (batch) for A; N-within-tile for B
    const int hi  = lane >> 4;         // selects K offset +0 or +2
    const float amask = (row < BSb) ? 1.f : 0.f;
    const int   r4    = row & 3;       // safe row for LDS read, masked to 0 if invalid
    const float* wb = fc1_w + (size_t)n * 320 * 64;

    for (int kk = 0; kk < 80; ++kk) {
        const int k0 = 4 * kk + 2 * hi;
        v2f av;
        av.x = amask * mi[r4 * 320 + k0];
        av.y = amask * mi[r4 * 320 + k0 + 1];
        const float* bp = wb + (size_t)k0 * 64 + row;
        v2f b0; b0.x = bp[0];  b0.y = bp[64];
        v2f b1; b1.x = bp[16]; b1.y = bp[80];
        v2f b2; b2.x = bp[32]; b2.y = bp[96];
        v2f b3; b3.x = bp[48]; b3.y = bp[112];
        acc0 = __builtin_amdgcn_wmma_f32_16x16x4_f32(false, av, false, b0, (short)0, acc0, false, false);
        acc1 = __builtin_amdgcn_wmma_f32_16x16x4_f32(false, av, false, b1, (short)0, acc1, false, false);
        acc2 = __builtin_amdgcn_wmma_f32_16x16x4_f32(false, av, false, b2, (short)0, acc2, false, false);
        acc3 = __builtin_amdgcn_wmma_f32_16x16x4_f32(false, av, false, b3, (short)0, acc3, false, false);
    }

    // ---- x = tanh(acc + bias). C/D layout: VGPR r, lanes 0-15 hold row M=r.
    if (lane < 16) {
        const float* bb = fc1_b + (size_t)n * 64;
        {
            int col = lane; float bias = bb[col];
            xs[0*64+col] = tanhf(acc0[0]+bias); xs[1*64+col] = tanhf(acc0[1]+bias);
            xs[2*64+col] = tanhf(acc0[2]+bias); xs[3*64+col] = tanhf(acc0[3]+bias);
        }
        {
            int col = 16 + lane; float bias = bb[col];
            xs[0*64+col] = tanhf(acc1[0]+bias); xs[1*64+col] = tanhf(acc1[1]+bias);
            xs[2*64+col] = tanhf(acc1[2]+bias); xs[3*64+col] = tanhf(acc1[3]+bias);
        }
        {
            int col = 32 + lane; float bias = bb[col];
            xs[0*64+col] = tanhf(acc2[0]+bias); xs[1*64+col] = tanhf(acc2[1]+bias);
            xs[2*64+col] = tanhf(acc2[2]+bias); xs[3*64+col] = tanhf(acc2[3]+bias);
        }
        {
            int col = 48 + lane; float bias = bb[col];
            xs[0*64+col] = tanhf(acc3[0]+bias); xs[1*64+col] = tanhf(acc3[1]+bias);
            xs[2*64+col] = tanhf(acc3[2]+bias); xs[3*64+col] = tanhf(acc3[3]+bias);
        }
    }
    __syncthreads();

    // ---- fc2: out3[b][o] (12 dot products of length 64)
    if (lane < 12) {
        int b = lane & 3, o = lane >> 2;
        float acc = fc2_b[(size_t)n * 3 + o];
        const float* w2 = fc2_w + (size_t)n * 64 * 3 + o;
        #pragma unroll 8
        for (int h = 0; h < 64; ++h) acc += xs[b * 64 + h] * w2[h * 3];
        o3[b * 3 + o] = acc;
    }
    __syncthreads();

    // ---- gates / decay / scale factors
    const float mlr = sigf(mod_lr_logit[0]);
    if (lane < 4) {
        int b = lane;
        float gp = tanhf(o3[b * 3 + 0]);
        float gk = tanhf(o3[b * 3 + 1]);
        float dm = o3[b * 3 + 2];
        eff_decay[(size_t)b * NNn + n] = sigf(decay_logit[n] + dm);
        sc[b]     = mlr * gp / fmaxf(nrm[b],     EPSF);
        sc[4 + b] = mlr * gk / fmaxf(nrm[4 + b], EPSF);
    }
    __syncthreads();

    // ---- eff_prim (to ws) and eff_key (to LDS)
    for (int i = lane; i < 256; i += 32) {
        int b = i >> 6, d = i & 63;
        float ep = mi[b * 320 + 192 + d] + sc[b]     * mi[b * 320 + 64  + d];
        float ev = mi[b * 320 + 256 + d] + sc[4 + b] * mi[b * 320 + 128 + d];
        eff_prim[((size_t)b * NNn + n) * 64 + d] = ep;
        ek[b * 64 + d] = ev;
    }
    __syncthreads();

    // ---- routing[b][k] = sigmoid(eff_key . prev_messages[b, conn[n,k]])
    for (int p = lane; p < BSb * KKk; p += 32) {
        int b = p >> 5, k = p & 31;
        int idx = conn[(size_t)n * KKk + k];
        const float* mrow = prev_msg + ((size_t)b * NNn + idx) * 64;
        float acc = 0.f;
        #pragma unroll 8
        for (int d = 0; d < 64; ++d) acc += ek[b * 64 + d] * mrow[d];
        routing[((size_t)b * NNn + n) * KKk + k] = sigf(acc);
    }
}

// ---------------------------------------------------------------------------
// Kernel 2: one scan step. One wave32 per (b, n); lane handles d and d+32.
// ---------------------------------------------------------------------------
__global__ void __launch_bounds__(256) k_step(
    const float* __restrict__ msg_in, const float* __restrict__ h_in,
    float* __restrict__ h_out, float* __restrict__ msg_out,
    const float* __restrict__ eff_prim, const float* __restrict__ eff_decay,
    const float* __restrict__ routing, const int* __restrict__ conn,
    const float* __restrict__ branch_w, const float* __restrict__ group_w,
    const float* __restrict__ cc, float* __restrict__ out, int t)
{
    const int wid  = (blockIdx.x * blockDim.x + threadIdx.x) >> 5;
    const int lane = threadIdx.x & 31;
    const int b = wid >> 12;           // N == 4096 == 2^12
    const int n = wid & (NNn - 1);

    float acc[4][2] = {};
    const int*   cn  = conn    + (size_t)n * KKk;
    const float* rt  = routing + ((size_t)b * NNn + n) * KKk;
    const float* bw0 = branch_w + (size_t)n * KKk * 64;

    for (int k = 0; k < KKk; ++k) {
        int idx = cn[k];
        float r = rt[k];
        const float* mrow = msg_in + ((size_t)b * NNn + idx) * 64;
        const float* bw = bw0 + k * 64;
        int j = k >> 3;
        acc[j][0] += r * mrow[lane]      * bw[lane];
        acc[j][1] += r * mrow[lane + 32] * bw[lane + 32];
    }

    float g0 = 0.f, g1 = 0.f;
    const float* gw0 = group_w + (size_t)n * 4 * 64;
    #pragma unroll
    for (int j = 0; j < 4; ++j) {
        const float* gw = gw0 + j * 64;
        g0 += tanhf(acc[j][0]) * gw[lane];
        g1 += tanhf(acc[j][1]) * gw[lane + 32];
    }
    float rec0 = tanhf(g0), rec1 = tanhf(g1);   // NG==1, mean is identity

    if (n < CCc) {
        const float* crow = cc + (((size_t)b * TSEG + 4 * t) * CCc + n) * 64;
        rec0 += crow[lane];
        rec1 += crow[lane + 32];
    }

    float ed = eff_decay[(size_t)b * NNn + n];
    size_t base = ((size_t)b * NNn + n) * 64;
    float h0 = ed * h_in[base + lane]      + (1.f - ed) * rec0;
    float h1 = ed * h_in[base + lane + 32] + (1.f - ed) * rec1;
    float m0 = tanhf(h0 * eff_prim[base + lane]);
    float m1 = tanhf(h1 * eff_prim[base + lane + 32]);
    h_out[base + lane]        = h0;
    h_out[base + lane + 32]   = h1;
    msg_out[base + lane]      = m0;
    msg_out[base + lane + 32] = m1;

    if (n < CCc) {
        #pragma unroll
        for (int i = 0; i < 4; ++i) {   // time-replicated ports
            size_t ob = (((size_t)b * TSEG + 4 * t + i) * CCc + n) * 64;
            out[ob + lane]      = m0;
            out[ob + lane + 32] = m1;
        }
    }
}

// ---------------------------------------------------------------------------
extern "C" void kernel_launch(void* const* d_in, const int* in_sizes, int n_in,
                              void* d_out, int out_size, void* d_ws, size_t ws_size,
                              hipStream_t stream) {
    const float* cc          = (const float*)d_in[0];
    const float* h_prev      = (const float*)d_in[1];
    const float* prev_msg    = (const float*)d_in[2];
    const float* trace_prim  = (const float*)d_in[3];
    const float* trace_key   = (const float*)d_in[4];
    const float* primitives  = (const float*)d_in[5];
    const float* key_p       = (const float*)d_in[6];
    const float* decay_logit = (const float*)d_in[7];
    const float* branch_w    = (const float*)d_in[8];
    const float* group_w     = (const float*)d_in[9];
    const float* fc1_w       = (const float*)d_in[10];
    const float* fc1_b       = (const float*)d_in[11];
    const float* fc2_w       = (const float*)d_in[12];
    const float* fc2_b       = (const float*)d_in[13];
    const float* mod_lr      = (const float*)d_in[14];
    const int*   conn        = (const int*)d_in[15];
    float* out = (float*)d_out;

    // Workspace layout (floats): ~19 MB total
    float* ws        = (float*)d_ws;
    float* eff_prim  = ws;                                  // 4*4096*64
    float* eff_decay = eff_prim + (size_t)BSb * NNn * DDd;  // 4*4096
    float* routing   = eff_decay + (size_t)BSb * NNn;       // 4*4096*32
    float* hbuf      = routing + (size_t)BSb * NNn * KKk;   // 4*4096*64
    float* msgA      = hbuf + (size_t)BSb * NNn * DDd;      // 4*4096*64
    float* msgB      = msgA + (size_t)BSb * NNn * DDd;      // 4*4096*64

    // Phase 1: modulator (one wave per node) — streams fc1_w once, WMMA GEMM
    k_mod<<<NNn / 8, 256, 0, stream>>>(h_prev, prev_msg, trace_prim, trace_key,
        primitives, key_p, decay_logit, fc1_w, fc1_b, fc2_w, fc2_b, mod_lr, conn,
        eff_prim, eff_decay, routing);

    // Phase 2: 4 scan steps; kernel boundaries provide global sync; msg ping-pong
    const int blocks = (BSb * NNn * 32) / 256;  // one wave per (b,n)
    k_step<<<blocks, 256, 0, stream>>>(prev_msg, h_prev, hbuf, msgA,
        eff_prim, eff_decay, routing, conn, branch_w, group_w, cc, out, 0);
    k_step<<<blocks, 256, 0, stream>>>(msgA, hbuf, hbuf, msgB,
        eff_prim, eff_decay, routing, conn, branch_w, group_w, cc, out, 1);
    k_step<<<blocks, 256, 0, stream>>>(msgB, hbuf, hbuf, msgA,
        eff_prim, eff_decay, routing, conn, branch_w, group_w, cc, out, 2);
    k_step<<<blocks, 256, 0, stream>>>(msgA, hbuf, hbuf, msgB,
        eff_prim, eff_decay, routing, conn, branch_w, group_w, cc, out, 3);
}